// TripletLoss_42150809043699
// MI455X (gfx1250) — compile-verified
//
#include <hip/hip_runtime.h>
#include <hip/hip_bf16.h>
#include <math.h>

typedef float v2f __attribute__((ext_vector_type(2)));
typedef float v8f __attribute__((ext_vector_type(8)));

#define NBLK   5
#define BATCH  1000
#define DIM    256
#define KSTEPS 64          // DIM / 4
#define TILES  63          // ceil(1000/16)
#define MARGIN 0.5f
#define EPS    1e-12f

// ---------------------------------------------------------------------------
// Kernel 1: gather the 5*1000 selected rows into the workspace, compute per-row
// squared norms and gather labels. One 256-thread block per output row.
// Permutation: affine bijection (j*7919 + uu) mod 100000 (gcd(7919,1e5)=1).
// ---------------------------------------------------------------------------
__global__ __launch_bounds__(DIM)
void gather_rows_kernel(const float* __restrict__ inputss,
                        const int*   __restrict__ targetss,
                        const int*   __restrict__ uu_ptr,
                        float*       __restrict__ Xg,     // [5*1000][256]
                        float*       __restrict__ sqw,    // [5*1000]
                        int*         __restrict__ tw)     // [5*1000]
{
    __shared__ float red[DIM];
    const int r   = blockIdx.x;          // 0..4999
    const int tid = threadIdx.x;         // 0..255
    const unsigned uu  = (unsigned)uu_ptr[0];
    const unsigned src = ((unsigned)r * 7919u + uu) % 100000u;

    const float v = inputss[(size_t)src * DIM + tid];
    Xg[(size_t)r * DIM + tid] = v;
    red[tid] = v * v;
    __syncthreads();
    #pragma unroll
    for (int s = DIM / 2; s > 0; s >>= 1) {
        if (tid < s) red[tid] += red[tid + s];
        __syncthreads();
    }
    if (tid == 0) {
        sqw[r] = red[0];
        tw[r]  = targetss[src];
    }
}

// ---------------------------------------------------------------------------
// Kernel 2: fused Gram-matrix (f32 WMMA) + distance + batch-hard mining + loss.
// grid = (63 row tiles, 5 blocks); 256 threads = 8 waves; wave w strides the
// column tiles ct = w, w+8, ...
//
// A panel (16 rows x 256 K) is preloaded once per wave into 128 VGPRs
// (64 x v2f, ISA 32-bit A layout: lane l holds row (l&15); K pair {0,1} for
// l<16, {2,3} for l>=16). Inner loop per column tile: 1 B load + 1 WMMA,
// dual accumulators for XDL ILP, prefetch of the next B panel (ct+8).
//
// C/D layout (32-bit, 16x16): VGPR v, lane l -> row M = v + 8*(l>=16),
//                             col N = (l & 15).
// ---------------------------------------------------------------------------
__global__ __launch_bounds__(256)
void triplet_wmma_kernel(const float* __restrict__ Xg,
                         const float* __restrict__ sqw,
                         const int*   __restrict__ tw,
                         float*       __restrict__ out)
{
    const int rt   = blockIdx.x;         // row tile 0..62
    const int b    = blockIdx.y;         // loss block 0..4
    const int tid  = threadIdx.x;
    const int lane = tid & 31;
    const int wave = tid >> 5;           // 0..7
    const int hi   = lane >> 4;          // upper half-wave?
    const int l15  = lane & 15;

    const int rowbase = rt * 16;

    // ---- Preload the A panel into registers (once per wave) ----
    const int arow = min(rowbase + l15, BATCH - 1);
    const float* Ap = Xg + ((size_t)b * BATCH + arow) * DIM + 2 * hi;
    v2f Areg[KSTEPS];
    #pragma unroll
    for (int kk = 0; kk < KSTEPS; ++kk)
        Areg[kk] = *(const v2f*)(Ap + kk * 4);

    // Per-lane metadata for the rows this lane owns in the C tile.
    float sqrow[8];
    int   trow[8];
    #pragma unroll
    for (int v = 0; v < 8; ++v) {
        const int rr = min(rowbase + v + 8 * hi, BATCH - 1);
        sqrow[v] = sqw[b * BATCH + rr];
        trow[v]  = tw[b * BATCH + rr];
    }

    float rmax[8], rmin[8];
    #pragma unroll
    for (int v = 0; v < 8; ++v) { rmax[v] = -INFINITY; rmin[v] = INFINITY; }

    for (int ct = wave; ct < TILES; ct += 8) {
        const int colbase = ct * 16;
        const int col     = colbase + l15;
        const int ccl     = min(col, BATCH - 1);
        const float* Bp   = Xg + ((size_t)b * BATCH + ccl) * DIM + 2 * hi;
        const float sqcol = sqw[b * BATCH + ccl];
        const int   tcol  = tw[b * BATCH + ccl];
        const bool  colok = (col < BATCH);

        // Prefetch this wave's next column-tile B panel (global_prefetch_b8).
        const int ctn = ct + 8;
        if (ctn < TILES) {
            const int cpn = min(ctn * 16 + l15, BATCH - 1);
            __builtin_prefetch(Xg + ((size_t)b * BATCH + cpn) * DIM, 0, 1);
        }

        // ---- Gram tile: 64 WMMAs, dual accumulators for ILP ----
        v8f c0 = {0.f, 0.f, 0.f, 0.f, 0.f, 0.f, 0.f, 0.f};
        v8f c1 = {0.f, 0.f, 0.f, 0.f, 0.f, 0.f, 0.f, 0.f};
        #pragma unroll
        for (int kk = 0; kk < KSTEPS; kk += 2) {
            v2f b0 = *(const v2f*)(Bp + kk * 4);
            v2f b1 = *(const v2f*)(Bp + kk * 4 + 4);
            // (neg_a, A, neg_b, B, c_mod, C, reuse_a, reuse_b)
            c0 = __builtin_amdgcn_wmma_f32_16x16x4_f32(
                     false, Areg[kk],     false, b0, (short)0, c0, false, false);
            c1 = __builtin_amdgcn_wmma_f32_16x16x4_f32(
                     false, Areg[kk + 1], false, b1, (short)0, c1, false, false);
        }

        // Fused distance + batch-hard mining on the 16x16 tile in registers.
        #pragma unroll
        for (int v = 0; v < 8; ++v) {
            const float g    = c0[v] + c1[v];
            const float d2   = sqrow[v] + sqcol - 2.0f * g;
            const float dist = __builtin_amdgcn_sqrtf(fmaxf(d2, EPS));
            const bool  same = (trow[v] == tcol);
            if (colok &&  same) rmax[v] = fmaxf(rmax[v], dist);
            if (colok && !same) rmin[v] = fminf(rmin[v], dist);
        }
    }

    // Reduce across the 16 lanes of each half-wave (xor stays within halves).
    #pragma unroll
    for (int v = 0; v < 8; ++v) {
        #pragma unroll
        for (int off = 8; off >= 1; off >>= 1) {
            rmax[v] = fmaxf(rmax[v], __shfl_xor(rmax[v], off, 32));
            rmin[v] = fminf(rmin[v], __shfl_xor(rmin[v], off, 32));
        }
    }

    // Combine the 8 waves' partial column-range results through LDS.
    __shared__ float smax[8][16];
    __shared__ float smin[8][16];
    if (l15 == 0) {
        #pragma unroll
        for (int v = 0; v < 8; ++v) {
            smax[wave][v + 8 * hi] = rmax[v];
            smin[wave][v + 8 * hi] = rmin[v];
        }
    }
    __syncthreads();

    if (tid < 16) {
        float m = -INFINITY, n = INFINITY;
        #pragma unroll
        for (int w = 0; w < 8; ++w) {
            m = fmaxf(m, smax[w][tid]);
            n = fminf(n, smin[w][tid]);
        }
        const int row = rowbase + tid;
        if (row < BATCH) {
            const float l = fmaxf(m - n + MARGIN, 0.0f);
            atomicAdd(out, l * (1.0f / (float)BATCH));   // mean over rows, sum over blocks
        }
    }
}

// ---------------------------------------------------------------------------
extern "C" void kernel_launch(void* const* d_in, const int* in_sizes, int n_in,
                              void* d_out, int out_size, void* d_ws, size_t ws_size,
                              hipStream_t stream) {
    (void)in_sizes; (void)n_in; (void)out_size; (void)ws_size;

    const float* inputss  = (const float*)d_in[0];   // [100000,256] f32
    const int*   targetss = (const int*)  d_in[1];   // [100000] i32
    const int*   uu       = (const int*)  d_in[2];   // scalar seed

    float* Xg  = (float*)d_ws;                       // 5*1000*256 f32 = 5.12 MB
    float* sqw = Xg + (size_t)NBLK * BATCH * DIM;    // 5*1000 f32
    int*   tw  = (int*)(sqw + NBLK * BATCH);         // 5*1000 i32
    float* out = (float*)d_out;

    hipMemsetAsync(out, 0, sizeof(float), stream);   // graph-capturable memset node

    gather_rows_kernel<<<NBLK * BATCH, DIM, 0, stream>>>(
        inputss, targetss, uu, Xg, sqw, tw);

    dim3 grid(TILES, NBLK);
    triplet_wmma_kernel<<<grid, 256, 0, stream>>>(Xg, sqw, tw, out);
}